// ProteinMPNNFeatures_87514253623873
// MI455X (gfx1250) — compile-verified
//
#include <hip/hip_runtime.h>
#include <hip/hip_bf16.h>

#define B_ 2
#define L_ 1024
#define K_ 30
#define NCH 416                 // 16 pos + 16 rbf(Dn) + 24*16 pair rbf
#define NEDGE (B_ * L_ * K_)    // 61440
#define LK (L_ * K_)

typedef __attribute__((ext_vector_type(2))) float v2f;
typedef __attribute__((ext_vector_type(8))) float v8f;

__constant__ int c_ia[24] = {0,2,3,4,1,1,1,1,0,0,0,4,4,3,0,2,3,4,2,3,4,2,3,2};
__constant__ int c_ib[24] = {0,2,3,4,0,2,3,4,2,3,4,2,3,2,1,1,1,1,0,0,0,4,4,3};

// ---------------------------------------------------------------------------
// Kernel 1: per-residue prep. atoms5[b,l,5,3] (N,Ca,C,O,Cb) and ridx[b,l].
// chain_idx is sorted per row with values in [0,4), so cumsum(boundary) ==
// popcount(present_mask below value). One block per batch row.
// ---------------------------------------------------------------------------
__global__ __launch_bounds__(1024) void prep_kernel(
    const float* __restrict__ X, const int* __restrict__ chain_idx,
    float* __restrict__ atoms5, int* __restrict__ ridx) {
  int b = blockIdx.x;
  int l = threadIdx.x;
  __shared__ unsigned smask;
  if (l == 0) smask = 0u;
  __syncthreads();
  int cv = chain_idx[b * L_ + l];
  atomicOr(&smask, 1u << cv);
  __syncthreads();
  unsigned mask = smask;
  int chain_num = __popc(mask & ((1u << cv) - 1u));
  int last = __popc(mask) - 1;
  ridx[b * L_ + l] = (chain_num < last) ? (100 * chain_num + l) : -100;

  const float* xp = X + (size_t)((b * L_ + l) * 4) * 3;
  float Nx = xp[0],  Ny = xp[1],  Nz = xp[2];
  float Cax = xp[3], Cay = xp[4], Caz = xp[5];
  float Cx = xp[6],  Cy = xp[7],  Cz = xp[8];
  float Ox = xp[9],  Oy = xp[10], Oz = xp[11];
  float bx = Cax - Nx, by = Cay - Ny, bz = Caz - Nz;
  float cx = Cx - Cax, cy = Cy - Cay, cz = Cz - Caz;
  float ax = by * cz - bz * cy;
  float ay = bz * cx - bx * cz;
  float az = bx * cy - by * cx;
  const float wa = -0.58273431f, wb = 0.56802827f, wc = -0.54067466f;
  float Cbx = wa * ax + wb * bx + wc * cx + Cax;
  float Cby = wa * ay + wb * by + wc * cy + Cay;
  float Cbz = wa * az + wb * bz + wc * cz + Caz;

  float* ap = atoms5 + (size_t)(b * L_ + l) * 15;
  ap[0] = Nx;  ap[1] = Ny;  ap[2] = Nz;
  ap[3] = Cax; ap[4] = Cay; ap[5] = Caz;
  ap[6] = Cx;  ap[7] = Cy;  ap[8] = Cz;
  ap[9] = Ox;  ap[10] = Oy; ap[11] = Oz;
  ap[12] = Cbx; ap[13] = Cby; ap[14] = Cbz;
}

// ---------------------------------------------------------------------------
// Kernel 2: RBF channels. One thread = (edge, group); group 0 -> rbf(Dn),
// groups 1..24 -> pair distances. 16 channels per thread, 4x float4 stores.
// ---------------------------------------------------------------------------
__global__ __launch_bounds__(256) void rbf_kernel(
    const float* __restrict__ atoms5, const int* __restrict__ E_idx,
    const float* __restrict__ D_neighbors, float* __restrict__ out) {
  int t = blockIdx.x * blockDim.x + threadIdx.x;
  if (t >= NEDGE * 25) return;
  int edge = t / 25;
  int g = t - edge * 25;
  int b = edge / LK;
  int rem = edge - b * LK;
  int l = rem / K_;

  float D;
  if (g == 0) {
    D = D_neighbors[edge];
  } else {
    int p = g - 1;
    int j = E_idx[edge];
    const float* A = atoms5 + (size_t)(b * L_ + l) * 15 + c_ia[p] * 3;
    const float* Bq = atoms5 + (size_t)(b * L_ + j) * 15 + c_ib[p] * 3;
    float dx = A[0] - Bq[0], dy = A[1] - Bq[1], dz = A[2] - Bq[2];
    D = sqrtf(dx * dx + dy * dy + dz * dz + 1e-6f);
  }

  float vals[16];
#pragma unroll
  for (int r = 0; r < 16; ++r) {
    float mu = 2.0f + (20.0f / 15.0f) * (float)r;  // linspace(2,22,16)
    float tt = (D - mu) * 0.8f;                    // sigma = 1.25
    vals[r] = __expf(-tt * tt);
  }
  float4* op = (float4*)(out + (size_t)edge * NCH + 16 + g * 16);  // 16B aligned
  op[0] = make_float4(vals[0], vals[1], vals[2], vals[3]);
  op[1] = make_float4(vals[4], vals[5], vals[6], vals[7]);
  op[2] = make_float4(vals[8], vals[9], vals[10], vals[11]);
  op[3] = make_float4(vals[12], vals[13], vals[14], vals[15]);
}

// ---------------------------------------------------------------------------
// Kernel 3: positional embedding via V_WMMA_F32_16X16X4_F32.
// D[m,n] = sum_K onehot(d_m)[K] * W_pos[n,K]  (bit-exact: 1.0*w + 0 adds).
// One wave per 16-edge tile; 61440/16 = 3840 tiles exactly -> EXEC all ones.
// A 16x4 f32 layout: VGPR0 = K{k0, k0+2}, VGPR1 = K{k0+1, k0+3} over lane
// halves; B 4x16 row-striped across lanes; C/D: VGPR v -> rows v / v+8.
// ---------------------------------------------------------------------------
__global__ __launch_bounds__(256) void pos_kernel(
    const int* __restrict__ chain_idx, const int* __restrict__ E_idx,
    const int* __restrict__ ridx, const float* __restrict__ W_pos,
    const float* __restrict__ b_pos, float* __restrict__ out) {
  int wave = (blockIdx.x * blockDim.x + threadIdx.x) >> 5;
  if (wave >= NEDGE / 16) return;  // uniform per wave; never taken at this grid
  int lane = threadIdx.x & 31;
  int m = lane & 15;
  int hi = lane >> 4;
  int e0 = wave * 16;
  int edge = e0 + m;

  int b = edge / LK;
  int rem = edge - b * LK;
  int l = rem / K_;
  int j = E_idx[edge];
  int off = ridx[b * L_ + l] - ridx[b * L_ + j] + 32;
  off = min(max(off, 0), 64);
  int same = (chain_idx[b * L_ + l] == chain_idx[b * L_ + j]);
  int d = same ? off : 65;
  int n = m;  // output channel handled by this lane

#if defined(__HIP_DEVICE_COMPILE__) && __has_builtin(__builtin_amdgcn_wmma_f32_16x16x4_f32)
  v8f acc = {};
#pragma unroll
  for (int k0 = 0; k0 < 68; k0 += 4) {   // K padded 66 -> 68 with zeros
    int ka = k0 + hi * 2;
    v2f A, Bm;
    A.x = (d == ka) ? 1.0f : 0.0f;
    A.y = (d == ka + 1) ? 1.0f : 0.0f;
    Bm.x = (ka < 66) ? W_pos[n * 66 + ka] : 0.0f;
    Bm.y = (ka + 1 < 66) ? W_pos[n * 66 + ka + 1] : 0.0f;
    acc = __builtin_amdgcn_wmma_f32_16x16x4_f32(
        /*neg_a=*/false, A, /*neg_b=*/false, Bm,
        /*c_mod=*/(short)0, acc, /*reuse_a=*/false, /*reuse_b=*/false);
  }
  float bp = b_pos[n];
#pragma unroll
  for (int v = 0; v < 8; ++v) {
    int row = v + hi * 8;
    out[(size_t)(e0 + row) * NCH + n] = acc[v] + bp;
  }
#else
  // Scalar fallback (host pass / missing builtin): direct one-hot gather.
  // hi=0 and hi=1 halves write identical values; benign.
#pragma unroll
  for (int c = 0; c < 16; ++c)
    out[(size_t)edge * NCH + c] = W_pos[c * 66 + d] + b_pos[c];
#endif
}

extern "C" void kernel_launch(void* const* d_in, const int* in_sizes, int n_in,
                              void* d_out, int out_size, void* d_ws, size_t ws_size,
                              hipStream_t stream) {
  const float* X           = (const float*)d_in[0];
  const int*   chain_idx   = (const int*)d_in[1];
  // d_in[2] = mask (unused by the computation)
  const int*   E_idx       = (const int*)d_in[3];
  const float* D_neighbors = (const float*)d_in[4];
  const float* W_pos       = (const float*)d_in[5];
  const float* b_pos       = (const float*)d_in[6];
  float* out = (float*)d_out;

  float* atoms5 = (float*)d_ws;                          // B*L*15 f32 = 120 KB
  int*   ridx   = (int*)((char*)d_ws + (size_t)B_ * L_ * 15 * sizeof(float));

  // Kernel 1: per-residue atoms (incl. Cb) + residue indices.
  prep_kernel<<<dim3(B_), dim3(L_), 0, stream>>>(X, chain_idx, atoms5, ridx);

  // Kernel 2: 400 RBF channels per edge (25 groups x 16 channels).
  {
    int total = NEDGE * 25;
    int blk = 256;
    rbf_kernel<<<dim3((total + blk - 1) / blk), dim3(blk), 0, stream>>>(
        atoms5, E_idx, D_neighbors, out);
  }

  // Kernel 3: 16 positional channels per edge via f32 WMMA (one wave / 16 edges).
  {
    int tiles = NEDGE / 16;          // 3840
    int waves_per_block = 256 / 32;  // 8
    int blocks = tiles / waves_per_block;  // 480, exact
    pos_kernel<<<dim3(blocks), dim3(256), 0, stream>>>(
        chain_idx, E_idx, ridx, W_pos, b_pos, out);
  }
}